// MixLoss_3315714752993
// MI455X (gfx1250) — compile-verified
//
#include <hip/hip_runtime.h>
#include <hip/hip_bf16.h>

// CDNA5 / gfx1250: fused L1 + SSIM loss.
// Separable 11-tap Gaussian depthwise conv done as banded matmuls on
// V_WMMA_F32_16X16X4_F32 (f32 precision, matrix pipe instead of VALU).

typedef float v2f __attribute__((ext_vector_type(2)));
typedef float v8f __attribute__((ext_vector_type(8)));

#define IMG_H 512
#define IMG_W 512
#define N_PLANES 96                       // 32 batch * 3 channels
#define N_TILES (N_PLANES * 32 * 32)      // 98304 16x16 tiles
#define WAVES_PER_WG 4
#define N_BLOCKS (N_TILES / WAVES_PER_WG) // 24576
#define N_TOTAL 25165824.0                // 32*3*512*512
#define SSIM_C1 1.0e-4f
#define SSIM_C2 9.0e-4f
#define ALPHA_ 0.84

#define EXT 26    // 16 + 2*5 halo
#define EXTP 28   // K padded to multiple of 4 for WMMA chunks

// Normalized 11-tap Gaussian, sigma = 1.5 (matches reference window)
__device__ __constant__ float GW[11] = {
    0.00102838f, 0.00759876f, 0.03600078f, 0.10936069f, 0.21300554f,
    0.26601173f, 0.21300554f, 0.10936069f, 0.03600078f, 0.00759876f,
    0.00102838f};

struct WaveBuf {
  float in1[EXT * EXTP];        // 728 f32
  float in2[EXT * EXTP];        // 728 f32
  float midp[5 * 14 * 16 * 2];  // [sig][kpair][n][{even,odd}] = 2240 f32
};
struct __align__(16) Smem {
  float wp[14 * 16 * 2];        // banded weight pairs [kpair][n][{even,odd}]
  WaveBuf wb[WAVES_PER_WG];
  float partial[WAVES_PER_WG][2];
};

__device__ __forceinline__ v8f wmma4(v2f a, v2f b, v8f c) {
  // D = A(16x4) x B(4x16) + C   (f32, wave32)
  return __builtin_amdgcn_wmma_f32_16x16x4_f32(false, a, false, b,
                                               (short)0, c, false, false);
}

__global__ __launch_bounds__(128) void ssim_l1_kernel(
    const float* __restrict__ pred, const float* __restrict__ targ,
    float* __restrict__ part) {
  __shared__ Smem sm;
  const int tid  = threadIdx.x;
  const int wv   = tid >> 5;
  const int lane = tid & 31;
  const int m    = lane & 15;  // A-row / B-col index of this lane
  const int hi   = lane >> 4;  // K half-select (ISA 7.12.2 f32 layouts)

  // Banded weight table: wp[pk][n] = { w[2pk-n], w[2pk+1-n] } (0 outside band).
  // Serves as B-fragment (Wh, pass 1) and A-fragment (Wv = Wh^T, pass 2).
  for (int i = tid; i < 14 * 16 * 2; i += 128) {
    int pk = i >> 5, n = (i >> 1) & 15, comp = i & 1;
    int d = 2 * pk + comp - n;
    sm.wp[i] = (d >= 0 && d <= 10) ? GW[d] : 0.0f;
  }

  const int tileIdx = blockIdx.x * WAVES_PER_WG + wv;
  const int plane = tileIdx >> 10;
  const int t = tileIdx & 1023;
  const int r0 = (t >> 5) << 4;
  const int c0 = (t & 31) << 4;
  const size_t pbase = (size_t)plane * (IMG_H * IMG_W);

  float* IN1  = sm.wb[wv].in1;
  float* IN2  = sm.wb[wv].in2;
  float* MIDP = sm.wb[wv].midp;

  // Load 26x26 halo tile, explicitly zero-padded (zero-pad conv semantics;
  // WMMA propagates NaN even through zero weights, so pads must be real 0).
  for (int idx = lane; idx < EXT * EXTP; idx += 32) {
    int rr = idx / EXTP, cc = idx - rr * EXTP;
    int gr = r0 - 5 + rr, gc = c0 - 5 + cc;
    bool ok = (cc < EXT) && (gr >= 0) && (gr < IMG_H) && (gc >= 0) && (gc < IMG_W);
    int cgr = min(max(gr, 0), IMG_H - 1);
    int cgc = min(max(gc, 0), IMG_W - 1);
    size_t off = pbase + (size_t)cgr * IMG_W + cgc;
    float a = pred[off];
    float b = targ[off];
    IN1[idx] = ok ? a : 0.0f;
    IN2[idx] = ok ? b : 0.0f;
  }
  // Zero K-pad rows 26,27 of Mid (pair index 13) for every signal.
  for (int s = 0; s < 5; ++s) MIDP[(s * 14 + 13) * 32 + lane] = 0.0f;

  __syncthreads();  // weight table ready for all waves

  // ---------- pass 1: horizontal conv.  Mid[r][c] = sum_j In[r][j]*w[j-c]
  // rows 0..25 covered by two 16-row groups (base 0 and 10; overlap rows are
  // rewritten with bit-identical values by the same wave -> benign).
  for (int g = 0; g < 2; ++g) {
    const int rb = g ? 10 : 0;
    v8f a0 = {0, 0, 0, 0, 0, 0, 0, 0};
    v8f a1 = a0, a2 = a0, a3 = a0, a4 = a0;
    const int arow = (rb + m) * EXTP;
#pragma unroll
    for (int ck = 0; ck < 7; ++ck) {
      const int k = ck * 4 + 2 * hi;
      v2f x1 = *(const v2f*)&IN1[arow + k];
      v2f x2 = *(const v2f*)&IN2[arow + k];
      v2f wb_ = *(const v2f*)&sm.wp[(k >> 1) * 32 + m * 2];
      a0 = wmma4(x1, wb_, a0);
      a1 = wmma4(x2, wb_, a1);
      a2 = wmma4(x1 * x1, wb_, a2);
      a3 = wmma4(x2 * x2, wb_, a3);
      a4 = wmma4(x1 * x2, wb_, a4);
    }
#pragma unroll
    for (int v = 0; v < 8; ++v) {  // D layout: VGPR v = rows rb+v / rb+v+8
      int row = rb + v + 8 * hi;
      int a = (row >> 1) * 32 + m * 2 + (row & 1);
      MIDP[0 * 448 + a] = a0[v];
      MIDP[1 * 448 + a] = a1[v];
      MIDP[2 * 448 + a] = a2[v];
      MIDP[3 * 448 + a] = a3[v];
      MIDP[4 * 448 + a] = a4[v];
    }
  }

  // ---------- pass 2: vertical conv.  Out = Wv x Mid, Wv = Wh^T (same table)
  v8f o0 = {0, 0, 0, 0, 0, 0, 0, 0};
  v8f o1 = o0, o2 = o0, o3 = o0, o4 = o0;
#pragma unroll
  for (int ck = 0; ck < 7; ++ck) {
    const int k = ck * 4 + 2 * hi;
    v2f wa = *(const v2f*)&sm.wp[(k >> 1) * 32 + m * 2];
    const int bb = (k >> 1) * 32 + m * 2;
    o0 = wmma4(wa, *(const v2f*)&MIDP[0 * 448 + bb], o0);
    o1 = wmma4(wa, *(const v2f*)&MIDP[1 * 448 + bb], o1);
    o2 = wmma4(wa, *(const v2f*)&MIDP[2 * 448 + bb], o2);
    o3 = wmma4(wa, *(const v2f*)&MIDP[3 * 448 + bb], o3);
    o4 = wmma4(wa, *(const v2f*)&MIDP[4 * 448 + bb], o4);
  }

  // ---------- SSIM map + L1, per-lane over 8 rows of the tile
  float l1s = 0.f, sss = 0.f;
#pragma unroll
  for (int v = 0; v < 8; ++v) {
    float mu1 = o0[v], mu2 = o1[v];
    float mu1s = mu1 * mu1, mu2s = mu2 * mu2, mu12 = mu1 * mu2;
    float s1 = o2[v] - mu1s;
    float s2 = o3[v] - mu2s;
    float s12 = o4[v] - mu12;
    float num = (2.f * mu12 + SSIM_C1) * (2.f * s12 + SSIM_C2);
    float den = (mu1s + mu2s + SSIM_C1) * (s1 + s2 + SSIM_C2);
    sss += num / den;
    int a = (5 + v + 8 * hi) * EXTP + 5 + m;  // tile center pixel
    l1s += fabsf(IN1[a] - IN2[a]);
  }

  // wave32 reduction
  for (int off = 16; off > 0; off >>= 1) {
    l1s += __shfl_xor(l1s, off);
    sss += __shfl_xor(sss, off);
  }
  if (lane == 0) { sm.partial[wv][0] = l1s; sm.partial[wv][1] = sss; }
  __syncthreads();
  if (tid == 0) {
    float a = 0.f, b = 0.f;
    for (int i = 0; i < WAVES_PER_WG; ++i) {
      a += sm.partial[i][0];
      b += sm.partial[i][1];
    }
    part[blockIdx.x * 2]     = a;   // L1 partial
    part[blockIdx.x * 2 + 1] = b;   // SSIM partial
  }
}

// Deterministic fixed-order final reduction (no atomics -> replay-stable).
__global__ __launch_bounds__(256) void finalize_kernel(
    const float* __restrict__ part, float* __restrict__ out) {
  __shared__ double sa[256], sb[256];
  double a = 0.0, b = 0.0;
  for (int i = threadIdx.x; i < N_BLOCKS; i += 256) {
    a += (double)part[2 * i];
    b += (double)part[2 * i + 1];
  }
  sa[threadIdx.x] = a;
  sb[threadIdx.x] = b;
  __syncthreads();
  for (int off = 128; off > 0; off >>= 1) {
    if (threadIdx.x < off) {
      sa[threadIdx.x] += sa[threadIdx.x + off];
      sb[threadIdx.x] += sb[threadIdx.x + off];
    }
    __syncthreads();
  }
  if (threadIdx.x == 0) {
    double inv = 1.0 / N_TOTAL;
    double l1 = sa[0] * inv;
    double ssim = sb[0] * inv;
    out[0] = (float)(ALPHA_ * l1 + (1.0 - ALPHA_) * (1.0 - ssim));
  }
}

extern "C" void kernel_launch(void* const* d_in, const int* in_sizes, int n_in,
                              void* d_out, int out_size, void* d_ws,
                              size_t ws_size, hipStream_t stream) {
  (void)in_sizes; (void)n_in; (void)out_size; (void)ws_size;
  const float* pred = (const float*)d_in[0];
  const float* targ = (const float*)d_in[1];
  float* part = (float*)d_ws;  // needs N_BLOCKS*2 floats = 192 KB scratch
  ssim_l1_kernel<<<N_BLOCKS, 128, 0, stream>>>(pred, targ, part);
  finalize_kernel<<<1, 256, 0, stream>>>(part, (float*)d_out);
}